// HuggarianMatcher_61177514164983
// MI455X (gfx1250) — compile-verified
//
#include <hip/hip_runtime.h>
#include <hip/hip_bf16.h>
#include <cstddef>

#define B_DIM 64
#define N_DIM 2048
#define M_DIM 64

typedef __attribute__((ext_vector_type(2))) float v2f;
typedef __attribute__((ext_vector_type(8))) float v8f;

// ---------------------------------------------------------------------------
// Phase 1: cost[b,n,m] = -ps[b,n,gs[b,m]] + sum_k |pb[b,n,k] - gb[b,m,k]|
// Presence term as f32 WMMA 16x16x4 (exact f32, K=2 zero-padded to 4):
//   A = ps tile (16x2), B = -onehot(gs) (2x16), D = presence (16x16).
// One wave per (b, 16-row n-tile); 4 WMMAs cover the 64 m columns.
// USET template param: when workspace is available we also write a transposed
// (B,M,N) copy (regular-temporal: it is re-read heavily by phase 2 and fits
// in the 192MB L2) and stream the (B,N,M) output with non-temporal stores
// (it is never re-read on device).
// ---------------------------------------------------------------------------
template <int USET>
__global__ __launch_bounds__(32) void cost_kernel(
    const float* __restrict__ ps,   // (B,N,2)
    const float* __restrict__ pb,   // (B,N,4)
    const int*   __restrict__ gs,   // (B,M)
    const float* __restrict__ gb,   // (B,M,4)
    float* __restrict__ cost,       // (B,N,M)
    float* __restrict__ costT)      // (B,M,N) scratch (USET=1 only)
{
  const int wid    = blockIdx.x;
  const int b      = wid / (N_DIM / 16);
  const int n_base = (wid % (N_DIM / 16)) * 16;
  const int lane   = threadIdx.x;
  const int hl     = lane >> 4;    // half-wave: K pair {0,1} vs {2,3}
  const int lo     = lane & 15;

  // A matrix 16x4 (MxK): lanes 0-15 -> rows, VGPR0/1 = K0/K1;
  // lanes 16-31 carry the zero K-padding (K2/K3).
  v2f a;
  if (hl == 0) {
    const float* p = ps + ((size_t)b * N_DIM + (n_base + lo)) * 2;
    a.x = p[0]; a.y = p[1];
  } else {
    a.x = 0.0f; a.y = 0.0f;
  }

  // Preload pb rows this lane contributes to in the D layout (rows r + 8*hl)
  float4 pbr[8];
#pragma unroll
  for (int r = 0; r < 8; ++r) {
    const int n = n_base + r + 8 * hl;
    pbr[r] = *(const float4*)(pb + ((size_t)b * N_DIM + n) * 4);
  }

#pragma unroll
  for (int mt = 0; mt < 4; ++mt) {
    const int m = mt * 16 + lo;

    // B matrix 4x16 (KxN): lanes 0-15 -> columns m with K0/K1 in two VGPRs;
    // lanes 16-31 hold the zero K-padding. Presence negation folded into B.
    v2f bm;
    if (hl == 0) {
      const int s = gs[b * M_DIM + m];
      bm.x = (s == 0) ? -1.0f : 0.0f;
      bm.y = (s == 1) ? -1.0f : 0.0f;
    } else {
      bm.x = 0.0f; bm.y = 0.0f;
    }

    v8f acc = {};
    acc = __builtin_amdgcn_wmma_f32_16x16x4_f32(
        /*neg_a=*/false, a, /*neg_b=*/false, bm,
        /*c_mod=*/(short)0, acc, /*reuse_a=*/false, /*reuse_b=*/false);

    const float4 g = *(const float4*)(gb + ((size_t)b * M_DIM + m) * 4);
#pragma unroll
    for (int r = 0; r < 8; ++r) {
      const int n = n_base + r + 8 * hl;   // D layout: row = r + 8*(lane>=16)
      const float l1 = fabsf(pbr[r].x - g.x) + fabsf(pbr[r].y - g.y) +
                       fabsf(pbr[r].z - g.z) + fabsf(pbr[r].w - g.w);
      const float val = acc[r] + l1;
      float* cdst = cost + ((size_t)b * N_DIM + n) * M_DIM + m;
      if (USET) {
        __builtin_nontemporal_store(val, cdst);   // dead after harness readback
        costT[((size_t)b * M_DIM + m) * N_DIM + n] = val;  // hot in L2 for phase 2
      } else {
        *cdst = val;                               // re-read by phase 2
      }
    }
  }
}

// ---------------------------------------------------------------------------
// Phase 2: assignment, replicating the reference's *actual* semantics.
// The reference never persists minv/way (mv/wy are numpy copies), so each
// inner iteration freshly argmins cost[i0-1,:] - u[i0] - v[:] over free
// columns, updates duals over used columns, and the augment collapses to
// p[j_final] = i (way is identically zero). One block per batch; the
// 2048-wide argmin is a strided scan + LDS tree reduction with numpy-style
// first-index tie-breaking.
// ---------------------------------------------------------------------------
__global__ __launch_bounds__(256) void hungarian_kernel(
    const float* __restrict__ cost,   // (B,N,M)
    const float* __restrict__ costT,  // (B,M,N) or null
    int useT,
    float* __restrict__ row_out,      // (B,M) as float
    float* __restrict__ col_out)      // (B,M) as float
{
  const int b  = blockIdx.x;
  const int t  = threadIdx.x;
  const int NT = 256;
  const float INF = __builtin_inff();

  __shared__ float v_lds[N_DIM + 1];
  __shared__ int   p_lds[N_DIM + 1];
  __shared__ unsigned char used[N_DIM + 1];
  __shared__ float u_lds[M_DIM + 1];
  __shared__ float red_val[256];
  __shared__ int   red_idx[256];
  __shared__ int   col4row[M_DIM];

  for (int j = t; j <= N_DIM; j += NT) { v_lds[j] = 0.0f; p_lds[j] = 0; }
  for (int i = t; i <= M_DIM; i += NT) u_lds[i] = 0.0f;
  __syncthreads();

  for (int i = 1; i <= M_DIM; ++i) {
    for (int j = t; j <= N_DIM; j += NT) used[j] = 0;
    if (t == 0) p_lds[0] = i;
    __syncthreads();

    int j0 = 0;
    for (;;) {
      if (t == 0) used[j0] = 1;
      __syncthreads();

      const int   i0  = p_lds[j0];
      const float ui0 = u_lds[i0];
      const float* crow =
          useT ? (costT + ((size_t)b * M_DIM + (i0 - 1)) * N_DIM) : nullptr;

      float bestv = INF;
      int   bestj = N_DIM + 1;
      for (int j = t + 1; j <= N_DIM; j += NT) {
        if (!used[j]) {
          const float c = useT
              ? crow[j - 1]
              : cost[((size_t)b * N_DIM + (j - 1)) * M_DIM + (i0 - 1)];
          const float cur = c - ui0 - v_lds[j];
          if (cur < bestv) { bestv = cur; bestj = j; }  // strict < keeps lowest j
        }
      }
      red_val[t] = bestv; red_idx[t] = bestj;
      __syncthreads();
      for (int s = NT / 2; s > 0; s >>= 1) {
        if (t < s) {
          const float v2 = red_val[t + s];
          const int   j2 = red_idx[t + s];
          if (v2 < red_val[t] || (v2 == red_val[t] && j2 < red_idx[t])) {
            red_val[t] = v2; red_idx[t] = j2;
          }
        }
        __syncthreads();
      }
      const float delta = red_val[0];
      const int   j1    = red_idx[0];

      // u[p[used]] += delta ; v[used] -= delta  (p values distinct: no conflicts)
      for (int j = t; j <= N_DIM; j += NT) {
        if (used[j]) { u_lds[p_lds[j]] += delta; v_lds[j] -= delta; }
      }
      __syncthreads();

      j0 = j1;
      if (p_lds[j0] == 0) break;
    }

    // way == 0 in the reference, so the augment chain is p[j_final] = p[0] = i
    if (t == 0) p_lds[j0] = i;
    __syncthreads();
  }

  // col4row[p[j]-1] = j-1
  for (int j = t + 1; j <= N_DIM; j += NT) {
    const int pi = p_lds[j];
    if (pi > 0) col4row[pi - 1] = j - 1;
  }
  __syncthreads();

  // argsort via rank counting (values are distinct): rows = sorted preds,
  // cols = argsort order
  if (t < M_DIM) {
    const int mine = col4row[t];
    int rank = 0;
    for (int k = 0; k < M_DIM; ++k) rank += (col4row[k] < mine);
    row_out[b * M_DIM + rank] = (float)mine;
    col_out[b * M_DIM + rank] = (float)t;
  }
}

extern "C" void kernel_launch(void* const* d_in, const int* in_sizes, int n_in,
                              void* d_out, int out_size, void* d_ws, size_t ws_size,
                              hipStream_t stream) {
  const float* ps = (const float*)d_in[0];  // (B,N,2) f32
  const float* pb = (const float*)d_in[1];  // (B,N,4) f32
  const int*   gs = (const int*)  d_in[2];  // (B,M)   i32
  const float* gb = (const float*)d_in[3];  // (B,M,4) f32

  float* out     = (float*)d_out;
  float* cost    = out;                                          // B*N*M
  float* row_out = out + (size_t)B_DIM * N_DIM * M_DIM;          // B*M
  float* col_out = row_out + (size_t)B_DIM * M_DIM;              // B*M

  const size_t needT = (size_t)B_DIM * M_DIM * N_DIM * sizeof(float);
  float* costT = (ws_size >= needT) ? (float*)d_ws : nullptr;
  const int useT = (costT != nullptr) ? 1 : 0;

  if (useT) {
    cost_kernel<1><<<B_DIM * (N_DIM / 16), 32, 0, stream>>>(
        ps, pb, gs, gb, cost, costT);
  } else {
    cost_kernel<0><<<B_DIM * (N_DIM / 16), 32, 0, stream>>>(
        ps, pb, gs, gb, cost, costT);
  }
  hungarian_kernel<<<B_DIM, 256, 0, stream>>>(
      cost, costT, useT, row_out, col_out);
}